// GCN_53145925321056
// MI455X (gfx1250) — compile-verified
//
#include <hip/hip_runtime.h>
#include <hip/hip_bf16.h>

typedef __attribute__((ext_vector_type(2))) float v2f;
typedef __attribute__((ext_vector_type(8))) float v8f;

#define HID 64

// ---------------------------------------------------------------------------
// degree / norm precompute
// ---------------------------------------------------------------------------
__global__ void k_init_deg(float* deg, int n) {
    int i = blockIdx.x * blockDim.x + threadIdx.x;
    if (i < n) deg[i] = 1.0f;                  // self-loop
}

__global__ void k_count_deg(const int* __restrict__ dst, float* deg, int e) {
    int i = blockIdx.x * blockDim.x + threadIdx.x;
    if (i < e) atomicAdd(&deg[dst[i]], 1.0f);
}

__global__ void k_deg_finalize(const float* __restrict__ deg,
                               float* __restrict__ dis,
                               float* __restrict__ dinv, int n) {
    int i = blockIdx.x * blockDim.x + threadIdx.x;
    if (i < n) {
        float d = deg[i];
        dis[i]  = rsqrtf(d);
        dinv[i] = 1.0f / d;
    }
}

__global__ void k_edge_norm(const int* __restrict__ src, const int* __restrict__ dst,
                            const float* __restrict__ dis, float* __restrict__ en, int e) {
    int i = blockIdx.x * blockDim.x + threadIdx.x;
    if (i < e) en[i] = dis[src[i]] * dis[dst[i]];
}

// ---------------------------------------------------------------------------
// WMMA fp32 GEMM: [rows,64] x [64,64], one 16x16 tile per wave (4 waves/block
// cover the 64 output columns).  mode 0: hout = A@W + bias (encoder)
//                               mode 1: hl = A@W ; hout = hl*dinv + bias
// ---------------------------------------------------------------------------
__global__ void k_gemm64_wmma(const float* __restrict__ A,    // [rows,64]
                              const float* __restrict__ W,    // [64,64]
                              const float* __restrict__ bias, // [64]
                              const float* __restrict__ dinv, // [rows] (mode 1)
                              float* __restrict__ hl,         // [rows,64] (mode 1)
                              float* __restrict__ hout,       // [rows,64]
                              int rows, int mode) {
    const int wave = threadIdx.x >> 5;        // 0..3 -> 16-col tile
    const int lane = threadIdx.x & 31;
    const int half = lane >> 4;               // K-pair selector for A/B frags
    const int lm   = lane & 15;
    const int row0 = blockIdx.x * 16;
    const int n0   = wave * 16;
    if (row0 >= rows) return;

    int arow = row0 + lm;
    if (arow >= rows) arow = rows - 1;        // clamp (rows%16==0 in practice)
    const float* Ap = A + (size_t)arow * HID;

    v8f acc = {};
#pragma unroll
    for (int kk = 0; kk < 16; ++kk) {
        const int kb = kk * 4 + half * 2;     // base K for this lane-half
        v2f a, b;
        a.x = Ap[kb];
        a.y = Ap[kb + 1];
        b.x = W[(size_t)kb * HID + n0 + lm];
        b.y = W[(size_t)(kb + 1) * HID + n0 + lm];
        acc = __builtin_amdgcn_wmma_f32_16x16x4_f32(
            /*neg_a=*/false, a, /*neg_b=*/false, b,
            /*c_mod=*/(short)0, acc, /*reuse_a=*/false, /*reuse_b=*/false);
    }

    const int   col = n0 + lm;
    const float bv  = bias[col];
#pragma unroll
    for (int v = 0; v < 8; ++v) {
        const int row = row0 + v + 8 * half;  // C layout: VGPR v, lane-half
        if (row >= rows) continue;
        const float val = acc[v];
        if (mode == 0) {
            hout[(size_t)row * HID + col] = val + bv;
        } else {
            hl  [(size_t)row * HID + col] = val;
            hout[(size_t)row * HID + col] = val * dinv[row] + bv; // self-loop + bias
        }
    }
}

// ---------------------------------------------------------------------------
// edge scatter: hnew[dst] += hl[src] * en   (16 threads/edge, 4 floats each)
// ---------------------------------------------------------------------------
__global__ void k_edge_scatter(const int* __restrict__ src, const int* __restrict__ dst,
                               const float* __restrict__ en, const float* __restrict__ hl,
                               float* __restrict__ hnew, int e) {
    long long t = (long long)blockIdx.x * blockDim.x + threadIdx.x;
    long long total = (long long)e * 16;
    if (t >= total) return;
    const int  ei = (int)(t >> 4);
    const int  fb = (int)(t & 15) * 4;
    const int  s  = src[ei];
    const int  d  = dst[ei];
    const float w = en[ei];
    const float4 m = *(const float4*)(hl + (size_t)s * HID + fb);
    float* dp = hnew + (size_t)d * HID + fb;
    atomicAdd(dp + 0, m.x * w);
    atomicAdd(dp + 1, m.y * w);
    atomicAdd(dp + 2, m.z * w);
    atomicAdd(dp + 3, m.w * w);
}

__global__ void k_silu(const float* __restrict__ hnew, float* __restrict__ h, int n) {
    int i = blockIdx.x * blockDim.x + threadIdx.x;
    if (i < n) {
        float x = hnew[i];
        h[i] = x / (1.0f + __expf(-x));
    }
}

// ---------------------------------------------------------------------------
// pooling + readout
// ---------------------------------------------------------------------------
__global__ void k_zero(float* p, int n) {
    int i = blockIdx.x * blockDim.x + threadIdx.x;
    if (i < n) p[i] = 0.0f;
}

__global__ void k_pool(const float* __restrict__ h, const int* __restrict__ batch,
                       float* __restrict__ pooled, int nodes) {
    long long t = (long long)blockIdx.x * blockDim.x + threadIdx.x;
    long long total = (long long)nodes * HID;
    if (t >= total) return;
    const int node = (int)(t >> 6);
    const int f    = (int)(t & 63);
    atomicAdd(&pooled[(size_t)batch[node] * HID + f], h[t]);
}

__global__ void k_readout(const float* __restrict__ pooled, const float* __restrict__ Wo,
                          const float* __restrict__ bo, float* __restrict__ out,
                          int graphs, int ncls) {
    int t = blockIdx.x * blockDim.x + threadIdx.x;
    if (t >= graphs * ncls) return;
    const int g = t / ncls;
    const int c = t % ncls;
    float s = bo[c];
    const float* pr = pooled + (size_t)g * HID;
#pragma unroll
    for (int k = 0; k < HID; ++k) s += pr[k] * Wo[(size_t)k * ncls + c];
    out[t] = s > 0.0f ? s : 0.0f;
}

// ---------------------------------------------------------------------------
// launcher
// ---------------------------------------------------------------------------
extern "C" void kernel_launch(void* const* d_in, const int* in_sizes, int n_in,
                              void* d_out, int out_size, void* d_ws, size_t ws_size,
                              hipStream_t stream) {
    const float* x      = (const float*)d_in[0];
    const int*   eidx   = (const int*)d_in[1];
    const int*   batch  = (const int*)d_in[2];
    const float* W_enc  = (const float*)d_in[3];
    const float* b_enc  = (const float*)d_in[4];
    const float* W_conv = (const float*)d_in[5];
    const float* b_conv = (const float*)d_in[6];
    const float* W_out  = (const float*)d_in[7];
    const float* b_out  = (const float*)d_in[8];

    const int N       = in_sizes[0] / HID;
    const int E       = in_sizes[1] / 2;
    const int LAYERS  = in_sizes[5] / (HID * HID);
    const int NCLS    = in_sizes[7] / HID;
    const int G       = out_size / NCLS;

    const int* src = eidx;          // edge_index[0]
    const int* dst = eidx + E;      // edge_index[1]

    // workspace carve-up (all offsets 16B aligned)
    char* ws = (char*)d_ws;
    size_t off = 0;
    float* deg  = (float*)(ws + off); off += (size_t)N * 4;
    float* dis  = (float*)(ws + off); off += (size_t)N * 4;
    float* dinv = (float*)(ws + off); off += (size_t)N * 4;
    float* en   = (float*)(ws + off); off += (size_t)E * 4;
    float* h    = (float*)(ws + off); off += (size_t)N * HID * 4;
    float* hl   = (float*)(ws + off); off += (size_t)N * HID * 4;
    float* hnew = (float*)(ws + off); off += (size_t)N * HID * 4;
    float* pooled = (float*)(ws + off); off += (size_t)G * HID * 4;

    const int T = 256;
    const int bN   = (N + T - 1) / T;
    const int bE   = (E + T - 1) / T;
    const int bNH  = (N * HID + T - 1) / T;
    const long long escT = (long long)E * 16;
    const int bESC = (int)((escT + T - 1) / T);
    const int gemmBlocks = (N + 15) / 16;

    // degree + edge norms
    k_init_deg     <<<bN, T, 0, stream>>>(deg, N);
    k_count_deg    <<<bE, T, 0, stream>>>(dst, deg, E);
    k_deg_finalize <<<bN, T, 0, stream>>>(deg, dis, dinv, N);
    k_edge_norm    <<<bE, T, 0, stream>>>(src, dst, dis, en, E);

    // encoder: h = x @ W_enc + b_enc   (WMMA fp32)
    k_gemm64_wmma<<<gemmBlocks, 128, 0, stream>>>(x, W_enc, b_enc, dinv,
                                                  hl, h, N, /*mode=*/0);

    for (int l = 0; l < LAYERS; ++l) {
        const float* Wl = W_conv + (size_t)l * HID * HID;
        const float* bl = b_conv + (size_t)l * HID;
        // hl = h @ Wl ; hnew = hl*dinv + bl   (WMMA fp32 + fused self-loop/bias)
        k_gemm64_wmma<<<gemmBlocks, 128, 0, stream>>>(h, Wl, bl, dinv,
                                                      hl, hnew, N, /*mode=*/1);
        // hnew[dst] += hl[src] * en
        k_edge_scatter<<<bESC, T, 0, stream>>>(src, dst, en, hl, hnew, E);
        // h = silu(hnew)
        k_silu<<<bNH, T, 0, stream>>>(hnew, h, N * HID);
    }

    // global add pool
    k_zero<<<(G * HID + T - 1) / T, T, 0, stream>>>(pooled, G * HID);
    k_pool<<<bNH, T, 0, stream>>>(h, batch, pooled, N);

    // readout
    k_readout<<<(G * NCLS + T - 1) / T, T, 0, stream>>>(pooled, W_out, b_out,
                                                        (float*)d_out, G, NCLS);
}